// KernelPlanes_89240830476844
// MI455X (gfx1250) — compile-verified
//
#include <hip/hip_runtime.h>
#include <math.h>

typedef float v2f __attribute__((ext_vector_type(2)));
typedef float v8f __attribute__((ext_vector_type(8)));

// D = A(16x4) * B(4x16) + C(16x16), all f32. 8-arg VOP3P form.
#define WMMA_F32(a, b, c) \
    __builtin_amdgcn_wmma_f32_16x16x4_f32(false, (a), false, (b), (short)0, (c), false, false)

// ---------------------------------------------------------------------------
// Prologue 1: transpose (C, S) channel-first tables to (S, C) channel-last so
// gathers become coalesced 128B wave loads (lane = channel).
// ---------------------------------------------------------------------------
__global__ void kp_transpose_CS(const float* __restrict__ src, float* __restrict__ dst,
                                int C, int S) {
    int i = blockIdx.x * blockDim.x + threadIdx.x;
    if (i < C * S) {
        int s = i / C;
        int c = i - s * C;
        dst[i] = src[c * S + s];   // dst[s*C + c] = src[c*S + s]
    }
}

// ---------------------------------------------------------------------------
// Prologue 2: pre-arrange weight matrices (K x N, row-major) into WMMA
// B-fragment order so each fragment is ONE coalesced global_load_b64:
//   frag[(kk*NT + nt)*32 + lane] = { W[kk*4 + 2*(lane/16) + 0][lane%16 + 16*nt],
//                                    W[kk*4 + 2*(lane/16) + 1][lane%16 + 16*nt] }
// ---------------------------------------------------------------------------
__global__ void kp_make_frags(const float* __restrict__ W, float* __restrict__ dst,
                              int K, int N, int NT) {
    int tid = blockIdx.x * blockDim.x + threadIdx.x;
    int nEntries = (K / 4) * NT * 32;
    if (tid >= nEntries) return;
    int kk  = tid / (NT * 32);
    int rem = tid - kk * (NT * 32);
    int nt  = rem >> 5;
    int l   = rem & 31;
    int hl  = l >> 4;
    int ml  = l & 15;
    int r   = kk * 4 + 2 * hl;
    int c   = ml + 16 * nt;
    dst[tid * 2 + 0] = W[r * N + c];
    dst[tid * 2 + 1] = W[(r + 1) * N + c];
}

// ---------------------------------------------------------------------------
// Fused kernel: 1 wave = 16 points. Gather -> LDS -> WMMA MLP pipeline.
//   sE:  6 matrices [16 pts][32 ch]   (row stride 34 to avoid bank conflicts)
//   sF:  [16 pts][40 ch] combined features (row stride 42)
// ---------------------------------------------------------------------------
#define E_STRIDE 34
#define E_SIZE   (16 * E_STRIDE)        // 544 floats per 16x32 matrix
#define F_STRIDE 42
#define F_SIZE   (16 * F_STRIDE)        // 672 floats
#define WAVE_LDS (6 * E_SIZE + F_SIZE)  // 3936 floats = 15744 B per wave

template <bool TR>
__global__ __launch_bounds__(128)
void kp_fused(const float* __restrict__ coords,
              const float* __restrict__ lx, const float* __restrict__ ly,
              const float* __restrict__ lz,
              const float* __restrict__ pxy, const float* __restrict__ pyz,
              const float* __restrict__ pxz,
              const float* __restrict__ vol,
              const float* __restrict__ Wx, const float* __restrict__ bx,
              const float* __restrict__ Wy, const float* __restrict__ by,
              const float* __restrict__ Wz, const float* __restrict__ bz,
              const float* __restrict__ W1, const float* __restrict__ b1,
              const float* __restrict__ W2, const float* __restrict__ b2,
              const float* __restrict__ fWx, const float* __restrict__ fWy,
              const float* __restrict__ fWz, const float* __restrict__ fW1,
              float* __restrict__ out, int M) {
    __shared__ float smem[4][WAVE_LDS];

    const int lane = threadIdx.x & 31;
    const int wid  = threadIdx.x >> 5;
    const int hl   = lane >> 4;      // half-wave: 0 or 1
    const int ml   = lane & 15;      // 0..15
    float* sE = smem[wid];
    float* sF = sE + 6 * E_SIZE;
    const int waveBase = (blockIdx.x * 4 + wid) * 16;

    // ---------------- Phase 1: gather (lane = channel) -----------------
    for (int p = 0; p < 16; ++p) {
        int pg = waveBase + p;
        if (pg >= M) pg = M - 1;                       // clamp (stores guarded later)
        const float cx = coords[pg * 3 + 0];
        const float cy = coords[pg * 3 + 1];
        const float cz = coords[pg * 3 + 2];

        auto samp1 = [&](const float* __restrict__ T, float g) -> float {
            float ix = (g + 1.0f) * 0.5f * 127.0f;
            float fx = floorf(ix);
            int   i0 = (int)fx;
            float w  = ix - fx;
            float r  = 0.0f;
#pragma unroll
            for (int d = 0; d < 2; ++d) {
                int   i  = i0 + d;
                float wt = d ? w : (1.0f - w);
                bool  v  = (i >= 0) && (i < 128);
                int   ic = i < 0 ? 0 : (i > 127 ? 127 : i);
                float val = TR ? T[ic * 32 + lane] : T[lane * 128 + ic];
                r += val * (v ? wt : 0.0f);
            }
            return r;
        };
        auto samp2 = [&](const float* __restrict__ T, float gx, float gy) -> float {
            float ix = (gx + 1.0f) * 0.5f * 127.0f;
            float iy = (gy + 1.0f) * 0.5f * 127.0f;
            float fx = floorf(ix), fy = floorf(iy);
            int   x0 = (int)fx,    y0 = (int)fy;
            float wx = ix - fx,    wy = iy - fy;
            float r = 0.0f;
#pragma unroll
            for (int dy = 0; dy < 2; ++dy)
#pragma unroll
                for (int dx = 0; dx < 2; ++dx) {
                    int   xi = x0 + dx, yi = y0 + dy;
                    float wt = (dx ? wx : 1.0f - wx) * (dy ? wy : 1.0f - wy);
                    bool  v  = (xi >= 0) && (xi < 128) && (yi >= 0) && (yi < 128);
                    int xc = xi < 0 ? 0 : (xi > 127 ? 127 : xi);
                    int yc = yi < 0 ? 0 : (yi > 127 ? 127 : yi);
                    int idx = yc * 128 + xc;
                    float val = TR ? T[idx * 32 + lane] : T[lane * 16384 + idx];
                    r += val * (v ? wt : 0.0f);
                }
            return r;
        };
        auto samp3 = [&](const float* __restrict__ T, float gx, float gy, float gz) -> float {
            float ix = (gx + 1.0f) * 0.5f * 63.0f;
            float iy = (gy + 1.0f) * 0.5f * 63.0f;
            float iz = (gz + 1.0f) * 0.5f * 63.0f;
            float fx = floorf(ix), fy = floorf(iy), fz = floorf(iz);
            int   x0 = (int)fx,    y0 = (int)fy,    z0 = (int)fz;
            float wx = ix - fx,    wy = iy - fy,    wz = iz - fz;
            float r = 0.0f;
            int c = lane & 7;
#pragma unroll
            for (int dz = 0; dz < 2; ++dz)
#pragma unroll
                for (int dy = 0; dy < 2; ++dy)
#pragma unroll
                    for (int dx = 0; dx < 2; ++dx) {
                        int xi = x0 + dx, yi = y0 + dy, zi = z0 + dz;
                        float wt = (dx ? wx : 1.0f - wx) * (dy ? wy : 1.0f - wy) *
                                   (dz ? wz : 1.0f - wz);
                        bool v = (xi >= 0) && (xi < 64) && (yi >= 0) && (yi < 64) &&
                                 (zi >= 0) && (zi < 64);
                        int xc = xi < 0 ? 0 : (xi > 63 ? 63 : xi);
                        int yc = yi < 0 ? 0 : (yi > 63 ? 63 : yi);
                        int zc = zi < 0 ? 0 : (zi > 63 ? 63 : zi);
                        int idx = (zc * 64 + yc) * 64 + xc;
                        float val = TR ? T[idx * 8 + c] : T[c * 262144 + idx];
                        r += val * (v ? wt : 0.0f);
                    }
            return r;
        };

        float xe  = samp1(lx, cx);
        float ye  = samp1(ly, cy);
        float ze  = samp1(lz, cz);
        float xye = samp2(pxy, cx, cy);
        float yze = samp2(pyz, cy, cz);
        float xze = samp2(pxz, cx, cz);
        float ve  = samp3(vol, cx, cy, cz);

        // relu(e) feeds the per-axis transforms; volume features stay raw.
        sE[0 * E_SIZE + p * E_STRIDE + lane] = fmaxf(xe, 0.0f);
        sE[1 * E_SIZE + p * E_STRIDE + lane] = fmaxf(ye, 0.0f);
        sE[2 * E_SIZE + p * E_STRIDE + lane] = fmaxf(ze, 0.0f);
        sE[3 * E_SIZE + p * E_STRIDE + lane] = fmaxf(yze, 0.0f);
        sE[4 * E_SIZE + p * E_STRIDE + lane] = fmaxf(xze, 0.0f);
        sE[5 * E_SIZE + p * E_STRIDE + lane] = fmaxf(xye, 0.0f);
        if (lane < 8) sF[p * F_STRIDE + 32 + lane] = ve;
    }
    __syncthreads();

    // ---------------- Phase 2: six 16x32 @ 32x32 transforms via WMMA ----
    // A fragment (f32 16x4): lane m%16 = row, VGPR v holds K = kk*4 + v + 2*(lane/16)
    // B fragment: TR path reads pre-arranged frags (one b64/lane, coalesced);
    //             fallback path assembles from the row-major weight matrix.
    auto phi = [&](const float* __restrict__ Es, const float* __restrict__ W,
                   const float* __restrict__ fW, const float* __restrict__ b,
                   v8f& d0, v8f& d1) {
        v8f a0 = {};
        v8f a1 = {};
#pragma unroll
        for (int kk = 0; kk < 8; ++kk) {
            int kbase = kk * 4 + 2 * hl;
            v2f A = *(const v2f*)&Es[ml * E_STRIDE + kbase];
            v2f B0, B1;
            if (TR) {
                B0 = *(const v2f*)&fW[((kk * 2 + 0) * 32 + lane) * 2];
                B1 = *(const v2f*)&fW[((kk * 2 + 1) * 32 + lane) * 2];
            } else {
                B0.x = W[kbase * 32 + ml];
                B0.y = W[(kbase + 1) * 32 + ml];
                B1.x = W[kbase * 32 + ml + 16];
                B1.y = W[(kbase + 1) * 32 + ml + 16];
            }
            a0 = WMMA_F32(A, B0, a0);
            a1 = WMMA_F32(A, B1, a1);
        }
        float bb0 = b[ml], bb1 = b[ml + 16];
#pragma unroll
        for (int v = 0; v < 8; ++v) { a0[v] += bb0; a1[v] += bb1; }
        d0 = a0;
        d1 = a1;
    };

    v8f f0, f1, t0, t1, u0, u1;
    phi(sE + 0 * E_SIZE, Wx, fWx, bx, t0, t1);   // xtran
    phi(sE + 3 * E_SIZE, Wx, fWx, bx, u0, u1);   // yztran
#pragma unroll
    for (int v = 0; v < 8; ++v) { f0[v] = t0[v] * u0[v]; f1[v] = t1[v] * u1[v]; }
    phi(sE + 1 * E_SIZE, Wy, fWy, by, t0, t1);   // ytran
    phi(sE + 4 * E_SIZE, Wy, fWy, by, u0, u1);   // xztran
#pragma unroll
    for (int v = 0; v < 8; ++v) { f0[v] += t0[v] * u0[v]; f1[v] += t1[v] * u1[v]; }
    phi(sE + 2 * E_SIZE, Wz, fWz, bz, t0, t1);   // ztran
    phi(sE + 5 * E_SIZE, Wz, fWz, bz, u0, u1);   // xytran
#pragma unroll
    for (int v = 0; v < 8; ++v) { f0[v] += t0[v] * u0[v]; f1[v] += t1[v] * u1[v]; }

    // D-fragment (16x16 f32): lanes 0-15 hold rows 0-7 across VGPRs, 16-31 rows 8-15.
#pragma unroll
    for (int v = 0; v < 8; ++v) {
        int m = v + 8 * hl;
        sF[m * F_STRIDE + ml]      = f0[v];
        sF[m * F_STRIDE + ml + 16] = f1[v];
    }
    __syncthreads();

    // ---------------- Phase 3: 16x40 @ 40x128 + bias + relu via WMMA ----
    v8f h[8];
#pragma unroll
    for (int nt = 0; nt < 8; ++nt) h[nt] = (v8f){};
#pragma unroll
    for (int kk = 0; kk < 10; ++kk) {
        int kbase = kk * 4 + 2 * hl;
        v2f A = *(const v2f*)&sF[ml * F_STRIDE + kbase];
#pragma unroll
        for (int nt = 0; nt < 8; ++nt) {
            v2f B;
            if (TR) {
                B = *(const v2f*)&fW1[((kk * 8 + nt) * 32 + lane) * 2];
            } else {
                int col = ml + 16 * nt;
                B.x = W1[kbase * 128 + col];
                B.y = W1[(kbase + 1) * 128 + col];
            }
            h[nt] = WMMA_F32(A, B, h[nt]);
        }
    }

    // ---------------- Phase 4: h @ W2 + b2 (128 -> 1) -------------------
    float part[8];
#pragma unroll
    for (int v = 0; v < 8; ++v) part[v] = 0.0f;
#pragma unroll
    for (int nt = 0; nt < 8; ++nt) {
        int col = ml + 16 * nt;
        float bb = b1[col];
        float w2 = W2[col];
#pragma unroll
        for (int v = 0; v < 8; ++v) {
            float hv = fmaxf(h[nt][v] + bb, 0.0f);
            part[v] += hv * w2;
        }
    }
    // reduce across the 16 lanes of each half-wave (rows 0-7 | 8-15)
#pragma unroll
    for (int v = 0; v < 8; ++v) {
        float s = part[v];
        s += __shfl_xor(s, 1, 32);
        s += __shfl_xor(s, 2, 32);
        s += __shfl_xor(s, 4, 32);
        s += __shfl_xor(s, 8, 32);
        part[v] = s;
    }
    float bias2 = b2[0];
    if (ml == 0) {            // lane 0 -> rows 0-7, lane 16 -> rows 8-15
#pragma unroll
        for (int v = 0; v < 8; ++v) {
            int pg = waveBase + v + 8 * hl;
            if (pg < M) out[pg] = part[v] + bias2;
        }
    }
}

// ---------------------------------------------------------------------------
extern "C" void kernel_launch(void* const* d_in, const int* in_sizes, int n_in,
                              void* d_out, int out_size, void* d_ws, size_t ws_size,
                              hipStream_t stream) {
    const float* coords = (const float*)d_in[0];
    const float* lx  = (const float*)d_in[1];
    const float* ly  = (const float*)d_in[2];
    const float* lz  = (const float*)d_in[3];
    const float* pxy = (const float*)d_in[4];
    const float* pyz = (const float*)d_in[5];
    const float* pxz = (const float*)d_in[6];
    const float* vol = (const float*)d_in[7];
    const float* Wx = (const float*)d_in[8];
    const float* bx = (const float*)d_in[9];
    const float* Wy = (const float*)d_in[10];
    const float* by = (const float*)d_in[11];
    const float* Wz = (const float*)d_in[12];
    const float* bz = (const float*)d_in[13];
    const float* W1 = (const float*)d_in[14];
    const float* b1 = (const float*)d_in[15];
    const float* W2 = (const float*)d_in[16];
    const float* b2 = (const float*)d_in[17];
    float* out = (float*)d_out;

    const int M = in_sizes[0] / 3;
    const int blocks = (M + 63) / 64;   // 4 waves x 16 points per block

    // Scratch layout (floats):
    //   3 lines (4096 ea) + 3 planes (524288 ea) + volume (2097152)
    //   + fragment weights: 3 x 1024 (32x32) + 5120 (40x128)
    const size_t LINE_F  = 32 * 128;      // 4096
    const size_t PLANE_F = 32 * 16384;    // 524288
    const size_t VOL_F   = 8 * 262144;    // 2097152
    const size_t FW_F    = 8 * 2 * 32 * 2;    // 1024 per 32x32 weight
    const size_t FW1_F   = 10 * 8 * 32 * 2;   // 5120 for 40x128 weight
    const size_t needF   = 3 * LINE_F + 3 * PLANE_F + VOL_F + 3 * FW_F + FW1_F;

    if (ws_size >= needF * sizeof(float)) {
        float* ws = (float*)d_ws;
        float* lxT  = ws;
        float* lyT  = lxT + LINE_F;
        float* lzT  = lyT + LINE_F;
        float* pxyT = lzT + LINE_F;
        float* pyzT = pxyT + PLANE_F;
        float* pxzT = pyzT + PLANE_F;
        float* volT = pxzT + PLANE_F;
        float* fWx  = volT + VOL_F;
        float* fWy  = fWx + FW_F;
        float* fWz  = fWy + FW_F;
        float* fW1  = fWz + FW_F;

        auto tp = [&](const float* s, float* d, int C, int S) {
            int n = C * S;
            kp_transpose_CS<<<(n + 255) / 256, 256, 0, stream>>>(s, d, C, S);
        };
        tp(lx, lxT, 32, 128);
        tp(ly, lyT, 32, 128);
        tp(lz, lzT, 32, 128);
        tp(pxy, pxyT, 32, 16384);
        tp(pyz, pyzT, 32, 16384);
        tp(pxz, pxzT, 32, 16384);
        tp(vol, volT, 8, 262144);

        auto mf = [&](const float* w, float* d, int K, int N, int NT) {
            int n = (K / 4) * NT * 32;
            kp_make_frags<<<(n + 255) / 256, 256, 0, stream>>>(w, d, K, N, NT);
        };
        mf(Wx, fWx, 32, 32, 2);
        mf(Wy, fWy, 32, 32, 2);
        mf(Wz, fWz, 32, 32, 2);
        mf(W1, fW1, 40, 128, 8);

        kp_fused<true><<<blocks, 128, 0, stream>>>(
            coords, lxT, lyT, lzT, pxyT, pyzT, pxzT, volT,
            Wx, bx, Wy, by, Wz, bz, W1, b1, W2, b2,
            fWx, fWy, fWz, fW1, out, M);
    } else {
        kp_fused<false><<<blocks, 128, 0, stream>>>(
            coords, lx, ly, lz, pxy, pyz, pxz, vol,
            Wx, bx, Wy, by, Wz, bz, W1, b1, W2, b2,
            nullptr, nullptr, nullptr, nullptr, out, M);
    }
    (void)n_in; (void)out_size;
}